// Attention2_5789615915322
// MI455X (gfx1250) — compile-verified
//
#include <hip/hip_runtime.h>
#include <hip/hip_bf16.h>

typedef float v2f __attribute__((ext_vector_type(2)));
typedef float v8f __attribute__((ext_vector_type(8)));

#define IN_CH  512
#define OUT_CH 512

// ---------------------------------------------------------------------------
// Kernel 0: zero the column-sum accumulators (ws is poisoned by the harness).
// ---------------------------------------------------------------------------
__global__ void k_zero(float* __restrict__ s) {
    int i = blockIdx.x * blockDim.x + threadIdx.x;
    if (i < 2 * OUT_CH) s[i] = 0.0f;
}

// ---------------------------------------------------------------------------
// Kernel 1: column sums of Z_l and Z_g.
//   256 threads/block: t<128 -> 4 cols of Z_l, t>=128 -> 4 cols of Z_g.
//   Coalesced float4 row reads, accumulate in regs, 4 atomicAdds at end.
// ---------------------------------------------------------------------------
__global__ void k_colsum(const float* __restrict__ Zl, const float* __restrict__ Zg,
                         float* __restrict__ sl, float* __restrict__ sg,
                         int N, int rowsPerBlk) {
    const int t = threadIdx.x;
    const bool isG = (t >= 128);
    const float* __restrict__ Z = isG ? Zg : Zl;
    float* __restrict__ s = isG ? sg : sl;
    const int c4 = t & 127;                 // which float4 of the 512-wide row

    int r0 = blockIdx.x * rowsPerBlk;
    int r1 = r0 + rowsPerBlk; if (r1 > N) r1 = N;

    float4 acc = make_float4(0.f, 0.f, 0.f, 0.f);
    for (int r = r0; r < r1; ++r) {
        const float4 v = *(const float4*)(Z + (size_t)r * OUT_CH + c4 * 4);
        acc.x += v.x; acc.y += v.y; acc.z += v.z; acc.w += v.w;
    }
    if (r0 < r1) {
        atomicAdd(&s[c4 * 4 + 0], acc.x);
        atomicAdd(&s[c4 * 4 + 1], acc.y);
        atomicAdd(&s[c4 * 4 + 2], acc.z);
        atomicAdd(&s[c4 * 4 + 3], acc.w);
    }
}

// ---------------------------------------------------------------------------
// Kernel 2: v_l = W^T s_l, v_g = W^T s_g (tiny 512x512 matvec, 1 block).
// Emits Q in WMMA-B-fragment step order:
//   Q[s][c][i] = V[4s + 2i + (c>>1)][c&1],  V[:,0]=v_l, V[:,1]=v_g
// so that lane-class c at K-step s reads its (b.x,b.y) pair as ONE b64:
//   flat offset = s*8 + c*2.
// Also c_l = b.s_l, c_g = b.s_g.
// ---------------------------------------------------------------------------
__global__ void k_makev(const float* __restrict__ W, const float* __restrict__ bvec,
                        const float* __restrict__ sl, const float* __restrict__ sg,
                        float* __restrict__ Q, float* __restrict__ cvec) {
    __shared__ float red[IN_CH];
    const int j = threadIdx.x;              // 512 threads, j = input channel k
    float vl = 0.f, vg = 0.f;
    for (int o = 0; o < OUT_CH; ++o) {
        const float w = W[(size_t)o * IN_CH + j];   // coalesced across j
        vl = fmaf(w, sl[o], vl);
        vg = fmaf(w, sg[o], vg);
    }
    // scatter into step-ordered Q:  k = 4s + r
    {
        const int s = j >> 2, r = j & 3;
        const int il = s * 8 + ((r < 2) ? (4 * r) : (4 * (r - 2) + 1));
        Q[il]     = vl;     // v_l[k] -> (c = 2*(r%2'), i)
        Q[il + 2] = vg;     // v_g[k] lands 2 floats later (next class)
    }

    // c_l = b . s_l
    red[j] = bvec[j] * sl[j];
    __syncthreads();
    for (int st = IN_CH / 2; st > 0; st >>= 1) {
        if (j < st) red[j] += red[j + st];
        __syncthreads();
    }
    if (j == 0) cvec[0] = red[0];
    __syncthreads();
    // c_g = b . s_g
    red[j] = bvec[j] * sg[j];
    __syncthreads();
    for (int st = IN_CH / 2; st > 0; st >>= 1) {
        if (j < st) red[j] += red[j + st];
        __syncthreads();
    }
    if (j == 0) cvec[1] = red[0];
}

// ---------------------------------------------------------------------------
// Kernel 3 (main, HBM-bound): per 16-row tile, one wave computes
//   att = Z_o_tile(16x512) @ V(512x16)   via 128x V_WMMA_F32_16X16X4_F32
// (V columns 0/1 = v_l/v_g, rest zero), then softmax-of-2 and
//   out = a0*Z_l + a1*Z_g  with coalesced b128 streaming.
//
// A-fragment (f32 16x4): lanes 0-15 row M=lane, regs = K{0,1}; lanes 16-31
// same rows, regs = K{2,3} -> one global b64 per lane per K-step.
// B-fragment: one unconditional ds_load_b64 per K-step from the step-ordered
// Q table; lanes outside {0,1,16,17} read a zeroed LDS region at the same
// stride (no EXEC games in the hot loop; EXEC stays all-1s for WMMA).
// Two independent accumulators over K-halves break the WMMA RAW chain.
// ---------------------------------------------------------------------------
__global__ void __launch_bounds__(256)
k_att_main(const float* __restrict__ Zo, const float* __restrict__ Zl,
           const float* __restrict__ Zg, const float* __restrict__ Q,
           const float* __restrict__ cvec, float* __restrict__ out, int N) {
    __shared__ float ldsQ[2048];            // 4 KB table + 4 KB zeros
    for (int i = threadIdx.x; i < 1024; i += blockDim.x) {
        ldsQ[i]        = Q[i];
        ldsQ[1024 + i] = 0.0f;
    }
    __syncthreads();

    const int lane = threadIdx.x & 31;
    const int wave = threadIdx.x >> 5;
    const int tile = blockIdx.x * (blockDim.x >> 5) + wave;
    const int nTiles = (N + 15) >> 4;
    if (tile >= nTiles) return;             // wave-uniform: EXEC stays full

    // A addressing: clamp OOB rows to a safe address; their att is garbage
    // but those rows are never written back.
    int rowA = tile * 16 + (lane & 15);
    if (rowA > N - 1) rowA = N - 1;
    const float* __restrict__ aptr = Zo + (size_t)rowA * IN_CH + ((lane >> 4) << 1);

    const bool act  = (lane & 0x0E) == 0;                 // lanes 0,1,16,17
    const int  pidx = ((lane >> 4) << 1) + (lane & 1);    // lane class 0..3
    const float* __restrict__ qlane = ldsQ + (act ? 0 : 1024) + pidx * 2;

    v8f acc0 = {};
    v8f acc1 = {};
#pragma unroll 4
    for (int s = 0; s < IN_CH / 8; ++s) {   // two K-steps per iteration
        // K-step s  (K = 4s)
        {
            const float2 av = *(const float2*)(aptr + 4 * s);
            v2f a; a.x = av.x; a.y = av.y;
            const v2f b = *(const v2f*)(qlane + 8 * s);
            acc0 = __builtin_amdgcn_wmma_f32_16x16x4_f32(
                false, a, false, b, (short)0, acc0, false, false);
        }
        // K-step s+64  (K = 256 + 4s)
        {
            const float2 av = *(const float2*)(aptr + 256 + 4 * s);
            v2f a; a.x = av.x; a.y = av.y;
            const v2f b = *(const v2f*)(qlane + 512 + 8 * s);
            acc1 = __builtin_amdgcn_wmma_f32_16x16x4_f32(
                false, a, false, b, (short)0, acc1, false, false);
        }
    }
    const v8f acc = acc0 + acc1;            // K-split recombine (8 VALU adds)

    const float cl = cvec[0];
    const float cg = cvec[1];

    // D layout: row m -> VGPR (m&7), lanes {0..15 | 16..31} for m {<8 | >=8};
    // att_l = column N=0, att_g = column N=1.
#pragma unroll
    for (int m = 0; m < 16; ++m) {
        const int src = (m < 8) ? 0 : 16;
        const float al = __shfl(acc[m & 7], src,     32) + cl;
        const float ag = __shfl(acc[m & 7], src + 1, 32) + cg;

        const int row = tile * 16 + m;
        if (row < N) {                       // wave-uniform
            const float mx = fmaxf(al, ag);
            const float el = __expf(al - mx);
            const float eg = __expf(ag - mx);
            const float inv = 1.0f / (el + eg);
            const float a0 = el * inv;
            const float a1 = eg * inv;

            const float4* __restrict__ zl4 = (const float4*)(Zl + (size_t)row * OUT_CH);
            const float4* __restrict__ zg4 = (const float4*)(Zg + (size_t)row * OUT_CH);
            float4* __restrict__ o4 = (float4*)(out + (size_t)row * OUT_CH);
#pragma unroll
            for (int it = 0; it < 4; ++it) {
                const int j = it * 32 + lane;
                const float4 l = zl4[j];
                const float4 g = zg4[j];
                float4 r;
                r.x = a0 * l.x + a1 * g.x;
                r.y = a0 * l.y + a1 * g.y;
                r.z = a0 * l.z + a1 * g.z;
                r.w = a0 * l.w + a1 * g.w;
                o4[j] = r;
            }
        }
    }
}

// ---------------------------------------------------------------------------
// Launch
// ---------------------------------------------------------------------------
extern "C" void kernel_launch(void* const* d_in, const int* in_sizes, int n_in,
                              void* d_out, int out_size, void* d_ws, size_t ws_size,
                              hipStream_t stream) {
    const float* Zo = (const float*)d_in[0];
    const float* Zl = (const float*)d_in[1];
    const float* Zg = (const float*)d_in[2];
    const float* W  = (const float*)d_in[3];
    const float* bv = (const float*)d_in[4];
    float* out = (float*)d_out;

    const int N = in_sizes[0] / IN_CH;

    float* ws   = (float*)d_ws;
    float* sl   = ws;                 // [512]
    float* sg   = ws + OUT_CH;        // [512]
    float* Q    = ws + 2 * OUT_CH;    // [1024] step-ordered B table
    float* cvec = ws + 4 * OUT_CH;    // [2]

    // 0: zero accumulators
    k_zero<<<(2 * OUT_CH + 255) / 256, 256, 0, stream>>>(ws);

    // 1: column sums of Z_l, Z_g
    const int nblk = 512;
    const int rowsPerBlk = (N + nblk - 1) / nblk;
    k_colsum<<<nblk, 256, 0, stream>>>(Zl, Zg, sl, sg, N, rowsPerBlk);

    // 2: v = W^T s (step-ordered), c = b.s
    k_makev<<<1, IN_CH, 0, stream>>>(W, bv, sl, sg, Q, cvec);

    // 3: WMMA att + softmax + combine
    const int nTiles = (N + 15) / 16;
    const int wavesPerBlk = 8;                       // 256 threads
    const int nb = (nTiles + wavesPerBlk - 1) / wavesPerBlk;
    k_att_main<<<nb, 32 * wavesPerBlk, 0, stream>>>(Zo, Zl, Zg, Q, cvec, out, N);
}